// QModel_65481071407675
// MI455X (gfx1250) — compile-verified
//
#include <hip/hip_runtime.h>

// CDNA5 / gfx1250 fused fraud-net forward.
// Memory-bound: 256 MB of x streamed once; pre-GEMM [B,64]@[64,2] done with
// V_WMMA_F32_16X16X4_F32; tiny tanh chain done per-lane after an LDS transpose.
// Inner loop steady state: 2x global_load_b64 + 2x v_wmma per K-chunk
// (B-fragments hoisted into registers before the loop).

typedef float v2f __attribute__((ext_vector_type(2)));
typedef float v8f __attribute__((ext_vector_type(8)));

#define NWAVES         8
#define ROWS_PER_WAVE  32
#define ROWS_PER_BLOCK (NWAVES * ROWS_PER_WAVE)   // 256
#define NF             64
#define NLAYERS        8

__device__ __forceinline__ float fast_tanh(float v) {
    // tanh(v) = 1 - 2/(e^{2v}+1);  e^{2v} = exp2(v * 2*log2(e))
    float e = __builtin_amdgcn_exp2f(v * 2.885390081777927f); // v_exp_f32
    return 1.0f - 2.0f * __builtin_amdgcn_rcpf(e + 1.0f);     // v_rcp_f32
}

__global__ __launch_bounds__(ROWS_PER_BLOCK)
void fraud_fused_wmma_kernel(const float* __restrict__ x,        // [B, 64]
                             const float* __restrict__ pre_w,    // [2, 64]
                             const float* __restrict__ pre_b,    // [2]
                             const float* __restrict__ layer_w,  // [8, 2, 2]
                             const float* __restrict__ layer_b,  // [8, 2]
                             const float* __restrict__ scale,    // [8, 2]
                             const float* __restrict__ shift,    // [8, 2]
                             const float* __restrict__ final_w,  // [1, 2]
                             const float* __restrict__ final_b,  // [1]
                             const float* __restrict__ head_w,   // [1, 1]
                             const float* __restrict__ head_b,   // [1]
                             float* __restrict__ out)            // [B]
{
    __shared__ float hbuf[NWAVES][ROWS_PER_WAVE * 2];

    const int tid  = threadIdx.x;
    const int wave = tid >> 5;
    const int lane = tid & 31;
    const int l16  = lane & 15;
    const int kh   = (lane >> 4) << 1;          // 0 or 2: K sub-offset per lane half

    const long long waveRow0 =
        (long long)blockIdx.x * ROWS_PER_BLOCK + (long long)wave * ROWS_PER_WAVE;

    // A-matrix feeds: two row groups of 16 (rows waveRow0.. and waveRow0+16..)
    const float* xa = x + (waveRow0 + l16)      * NF + kh;
    const float* xb = x + (waveRow0 + 16 + l16) * NF + kh;

    // ---- Hoist B-matrix fragments (loop-invariant weights) into registers.
    // B[k][n] = pre_w[n][k] for n<2, else 0.   n = lane&15.
    // Fragment layout per chunk c (K base = 4c): vgpr0 = K=4c+kh, vgpr1 = K=4c+kh+1.
    const bool nvalid = (l16 < 2);
    const v2f* pw2 = (const v2f*)(pre_w + (l16 & 1) * NF + kh);  // 8B-aligned (kh even)
    v2f bfrag[16];
#pragma unroll
    for (int c = 0; c < 16; ++c) {
        v2f w = pw2[c * 2];                     // pre_w[n][4c+kh], pre_w[n][4c+kh+1]
        bfrag[c].x = nvalid ? w.x : 0.0f;       // v_cndmask, EXEC untouched
        bfrag[c].y = nvalid ? w.y : 0.0f;
    }

    v8f accA = {};
    v8f accB = {};

#pragma unroll
    for (int c = 0; c < 16; ++c) {              // K = 64 in chunks of 4
        const int k0 = c * 4;
        v2f a0, a1;
        a0.x = xa[k0];  a0.y = xa[k0 + 1];      // global_load_b64, streamed from HBM
        a1.x = xb[k0];  a1.y = xb[k0 + 1];
        // D = A(16x4 f32) x B(4x16 f32) + C   -> v_wmma_f32_16x16x4_f32
        accA = __builtin_amdgcn_wmma_f32_16x16x4_f32(
                   false, a0, false, bfrag[c], (short)0, accA, false, false);
        accB = __builtin_amdgcn_wmma_f32_16x16x4_f32(
                   false, a1, false, bfrag[c], (short)0, accB, false, false);
    }

    // Scatter the two useful columns (N=0,1) of each 16x16 D tile to LDS so
    // every lane can own one batch row for the elementwise chain.
    if (l16 < 2) {
        const int n     = l16;
        const int mbase = (lane >> 4) * 8;      // D layout: vgpr r -> M = r + 8*(lane/16)
#pragma unroll
        for (int r = 0; r < 8; ++r) {
            hbuf[wave][(mbase + r) * 2 + n]      = accA[r];
            hbuf[wave][(16 + mbase + r) * 2 + n] = accB[r];
        }
    }
    __syncthreads();

    // Per-lane tail: row = waveRow0 + lane
    float h0 = hbuf[wave][lane * 2 + 0] + pre_b[0];
    float h1 = hbuf[wave][lane * 2 + 1] + pre_b[1];
    h0 = fmaxf(h0, 0.0f);
    h1 = fmaxf(h1, 0.0f);

#pragma unroll
    for (int l = 0; l < NLAYERS; ++l) {         // params are lane-uniform -> s_load
        const float w00 = layer_w[l * 4 + 0], w01 = layer_w[l * 4 + 1];
        const float w10 = layer_w[l * 4 + 2], w11 = layer_w[l * 4 + 3];
        float t0 = fast_tanh(fmaf(w00, h0, fmaf(w01, h1, layer_b[l * 2 + 0])));
        float t1 = fast_tanh(fmaf(w10, h0, fmaf(w11, h1, layer_b[l * 2 + 1])));
        h0 = fmaf(t0, scale[l * 2 + 0], shift[l * 2 + 0]);
        h1 = fmaf(t1, scale[l * 2 + 1], shift[l * 2 + 1]);
    }

    float f = fmaf(h0, final_w[0], fmaf(h1, final_w[1], final_b[0]));
    out[waveRow0 + lane] = fmaf(f, head_w[0], head_b[0]);
}

extern "C" void kernel_launch(void* const* d_in, const int* in_sizes, int n_in,
                              void* d_out, int out_size, void* d_ws, size_t ws_size,
                              hipStream_t stream) {
    const float* x       = (const float*)d_in[0];
    const float* pre_w   = (const float*)d_in[1];
    const float* pre_b   = (const float*)d_in[2];
    const float* layer_w = (const float*)d_in[3];
    const float* layer_b = (const float*)d_in[4];
    const float* scale   = (const float*)d_in[5];
    const float* shift   = (const float*)d_in[6];
    const float* final_w = (const float*)d_in[7];
    const float* final_b = (const float*)d_in[8];
    const float* head_w  = (const float*)d_in[9];
    const float* head_b  = (const float*)d_in[10];
    float* out = (float*)d_out;

    const int Bn = in_sizes[0] / NF;            // 1<<20; divisible by 256 by construction
    dim3 grid(Bn / ROWS_PER_BLOCK);
    dim3 block(ROWS_PER_BLOCK);
    fraud_fused_wmma_kernel<<<grid, block, 0, stream>>>(
        x, pre_w, pre_b, layer_w, layer_b, scale, shift,
        final_w, final_b, head_w, head_b, out);
}